// HalfKPInputLayer_43490838839494
// MI455X (gfx1250) — compile-verified
//
#include <hip/hip_runtime.h>
#include <hip/hip_bf16.h>

// ---------------------------------------------------------------------------
// HalfKP input layer on MI455X (gfx1250), wave32.
//
// Strategy: bin the 2048 (batch, perspective) pairs by king square (64 bins),
// then each bin is a dense GEMM  P(n_k x 640) * W[k](640 x 256)  computed with
// V_WMMA_F32_16X16X4_F32 tiles (fp32 in/out -> matches fp32 reference math).
// Problem is L2-bandwidth bound (42MB weight table << 192MB L2), so the fp32
// WMMA path is both exact and nowhere near the matrix-unit ceiling.
// ---------------------------------------------------------------------------

typedef __attribute__((ext_vector_type(2))) float v2f;
typedef __attribute__((ext_vector_type(8))) float v8f;

#define NB    1024              // batch
#define NPAIR 2048              // batch * 2 perspectives
#define NF    640               // features
#define NC    256               // channels
#define NK    64                // king squares
#define MAXT  192               // upper bound on padded 16-row M-tiles

// ---- phase 1: histogram of (b,s) pairs per king ---------------------------
__global__ void halfkp_zero_counts(int* __restrict__ counts) {
    if (threadIdx.x < NK) counts[threadIdx.x] = 0;
}

__global__ void halfkp_count(const int* __restrict__ kings, int* __restrict__ counts) {
    int t = blockIdx.x * blockDim.x + threadIdx.x;
    if (t < NPAIR) atomicAdd(&counts[kings[t] & (NK - 1)], 1);
}

// ---- phase 2: serial scan -> bin offsets + padded tile table --------------
__global__ void halfkp_scan(const int* __restrict__ counts,
                            int* __restrict__ offsets, int* __restrict__ cursors,
                            int* __restrict__ tileKing, int* __restrict__ tileBase,
                            int* __restrict__ tileCnt, int* __restrict__ nTiles) {
    if (threadIdx.x != 0 || blockIdx.x != 0) return;
    int off = 0, t = 0;
    for (int k = 0; k < NK; ++k) {
        offsets[k] = off;
        cursors[k] = 0;
        int c  = counts[k];
        int nt = (c + 15) >> 4;
        for (int i = 0; i < nt && t < MAXT; ++i) {
            tileKing[t] = k;
            tileBase[t] = off + i * 16;
            int rem = c - i * 16;
            tileCnt[t]  = rem < 16 ? rem : 16;
            ++t;
        }
        off += c;
    }
    nTiles[0] = t;
}

// ---- phase 3: scatter pair ids into bins ----------------------------------
__global__ void halfkp_scatter(const int* __restrict__ kings,
                               const int* __restrict__ offsets,
                               int* __restrict__ cursors, int* __restrict__ pairIdx) {
    int t = blockIdx.x * blockDim.x + threadIdx.x;
    if (t < NPAIR) {
        int k   = kings[t] & (NK - 1);
        int pos = offsets[k] + atomicAdd(&cursors[k], 1);
        pairIdx[pos] = t;
    }
}

// ---- phase 4: per-king GEMM with fp32 WMMA --------------------------------
// grid = (MAXT, 4), block = 32 (one wave). blockIdx.y picks a 64-column slab.
__global__ void halfkp_gemm(const unsigned char* __restrict__ pp,   // (B,640) bool
                            const float* __restrict__ W,            // (64,641,256)
                            const int* __restrict__ pairIdx,
                            const int* __restrict__ tileKing,
                            const int* __restrict__ tileBase,
                            const int* __restrict__ tileCnt,
                            const int* __restrict__ nTiles,
                            float* __restrict__ out2) {             // (2048,256)
    int tile = blockIdx.x;
    if (tile >= nTiles[0]) return;                 // uniform per block

    const int k    = tileKing[tile];
    const int base = tileBase[tile];
    const int cnt  = tileCnt[tile];

    const int lane  = threadIdx.x & 31;
    const int rowid = lane & 15;
    const int hi    = lane >> 4;                   // lane group 0/1
    const int klo   = hi << 1;                     // K sub-offset: 0 or 2

    // A operand row for this lane (ISA layout: lanes0-15 K=0,1; lanes16-31 K=2,3)
    const bool aValid = rowid < cnt;
    const int  aPair  = pairIdx[base + (aValid ? rowid : 0)];
    const unsigned char* pA = pp + (size_t)(aPair >> 1) * NF + klo;
    const float amask = aValid ? 1.0f : 0.0f;

    const float* Wk = W + (size_t)k * (NF + 1) * NC;
    const int c0 = blockIdx.y * 64;                // this wave's 64 columns

    v8f acc0 = {}, acc1 = {}, acc2 = {}, acc3 = {};

    for (int f0 = 0; f0 < NF; f0 += 4) {
        // A: 16x4 fp32 tile of 0/1 features (masked for padded rows)
        v2f a;
        a.x = amask * (float)pA[f0];
        a.y = amask * (float)pA[f0 + 1];

        // B: 4x16 fp32 weight tiles; x = row f0+klo, y = row f0+klo+1
        const float* Wf = Wk + (size_t)(f0 + klo) * NC;
        const int col = c0 + rowid;
        v2f b0; b0.x = Wf[col];      b0.y = Wf[NC + col];
        v2f b1; b1.x = Wf[col + 16]; b1.y = Wf[NC + col + 16];
        v2f b2; b2.x = Wf[col + 32]; b2.y = Wf[NC + col + 32];
        v2f b3; b3.x = Wf[col + 48]; b3.y = Wf[NC + col + 48];

        acc0 = __builtin_amdgcn_wmma_f32_16x16x4_f32(false, a, false, b0,
                                                     (short)0, acc0, false, false);
        acc1 = __builtin_amdgcn_wmma_f32_16x16x4_f32(false, a, false, b1,
                                                     (short)0, acc1, false, false);
        acc2 = __builtin_amdgcn_wmma_f32_16x16x4_f32(false, a, false, b2,
                                                     (short)0, acc2, false, false);
        acc3 = __builtin_amdgcn_wmma_f32_16x16x4_f32(false, a, false, b3,
                                                     (short)0, acc3, false, false);
    }

    // Epilogue: D layout = 8 VGPRs; lanes0-15 -> M=r, lanes16-31 -> M=r+8,
    // N = lane&15. Add the king's "extra" row (feature 640) here; each (b,s)
    // pair is written by exactly one tile row -> plain deterministic stores.
    const int Mbase = hi << 3;
    const float* Wextra = Wk + (size_t)NF * NC;
    v8f accs[4] = {acc0, acc1, acc2, acc3};
#pragma unroll
    for (int t = 0; t < 4; ++t) {
        const int col = c0 + t * 16 + rowid;
        const float extra = Wextra[col];
#pragma unroll
        for (int r = 0; r < 8; ++r) {
            const int M = Mbase + r;
            if (M < cnt) {
                const int pr = pairIdx[base + M];
                out2[(size_t)pr * NC + col] = accs[t][r] + extra;
            }
        }
    }
}

// ---- phase 5: combine the two perspectives + bias -------------------------
__global__ void halfkp_reduce(const float* __restrict__ out2,
                              const float* __restrict__ bias,
                              float* __restrict__ out) {
    int i = blockIdx.x * blockDim.x + threadIdx.x;   // b*256 + c
    if (i < NB * NC) {
        int b = i >> 8, c = i & (NC - 1);
        out[i] = bias[c] + out2[(size_t)(2 * b) * NC + c]
                         + out2[(size_t)(2 * b + 1) * NC + c];
    }
}

extern "C" void kernel_launch(void* const* d_in, const int* in_sizes, int n_in,
                              void* d_out, int out_size, void* d_ws, size_t ws_size,
                              hipStream_t stream) {
    const unsigned char* pp = (const unsigned char*)d_in[0];  // bool (B,64,10)
    const int*   kings = (const int*)d_in[1];                 // (B,2) int32
    const float* W     = (const float*)d_in[2];               // (64,641,256) f32
    const float* bias  = (const float*)d_in[3];               // (256,) f32
    float* out = (float*)d_out;

    // workspace layout (ints then the 2MB partial buffer, 16B aligned)
    int* ws       = (int*)d_ws;
    int* counts   = ws;                 // 64
    int* offsets  = ws + 64;            // 64
    int* cursors  = ws + 128;           // 64
    int* pairIdx  = ws + 192;           // 2048
    int* tileKing = ws + 2240;          // 192
    int* tileBase = ws + 2432;          // 192
    int* tileCnt  = ws + 2624;          // 192
    int* nTiles   = ws + 2816;          // 1 (+pad)
    float* out2   = (float*)(ws + 2832); // 2048*256 f32 = 2MB

    halfkp_zero_counts<<<1, 64, 0, stream>>>(counts);
    halfkp_count<<<NPAIR / 256, 256, 0, stream>>>(kings, counts);
    halfkp_scan<<<1, 1, 0, stream>>>(counts, offsets, cursors,
                                     tileKing, tileBase, tileCnt, nTiles);
    halfkp_scatter<<<NPAIR / 256, 256, 0, stream>>>(kings, offsets, cursors, pairIdx);
    halfkp_gemm<<<dim3(MAXT, 4), 32, 0, stream>>>(pp, W, pairIdx, tileKing,
                                                  tileBase, tileCnt, nTiles, out2);
    halfkp_reduce<<<(NB * NC) / 256, 256, 0, stream>>>(out2, bias, out);
}